// ScaledDotProductAttention_41171556500099
// MI455X (gfx1250) — compile-verified
//
#include <hip/hip_runtime.h>
#include <hip/hip_bf16.h>
#include <stdint.h>

typedef __attribute__((ext_vector_type(2))) float v2f;
typedef __attribute__((ext_vector_type(4))) float v4f;
typedef __attribute__((ext_vector_type(8))) float v8f;

#define BH 32
#define NN 256
#define DD 64
#define NEG_INF -1.0e9f

// ---------------------------------------------------------------------------
// Kernel 1: raw scores S = q @ k^T  per batch, via V_WMMA_F32_16X16X4_F32.
// One wave computes one 16x16 output tile; 8 waves (256 thr) per block.
// grid = BH * (N/16 i-tiles) * (N/16/8 j-groups) = 32*16*2 = 1024 blocks.
// ---------------------------------------------------------------------------
__global__ void qk_wmma_kernel(const float* __restrict__ q,
                               const float* __restrict__ k,
                               float* __restrict__ attn) {
    const int wave = threadIdx.x >> 5;
    const int lane = threadIdx.x & 31;
    const int half = lane >> 4;       // K-pair select (A/B layout halves)
    const int l    = lane & 15;       // M (for A) / N (for B) index

    const int blk = blockIdx.x;
    const int jg  = blk & 1;                 // 2 j-groups of 8 tiles
    const int it  = (blk >> 1) & 15;         // 16 i-tiles
    const int b   = blk >> 5;                // batch

    const int i0 = it * 16;
    const int j0 = (jg * 8 + wave) * 16;

    const float* __restrict__ qb = q + ((size_t)b * NN + i0) * DD;
    const float* __restrict__ kb = k + ((size_t)b * NN + j0) * DD;

    v8f c = {};
#pragma unroll
    for (int kk = 0; kk < DD; kk += 4) {
        // A: 16x4 tile of Q rows.  lane<16 -> K=kk,kk+1 ; lane>=16 -> K=kk+2,kk+3
        v2f a  = *(const v2f*)(qb + (size_t)l * DD + kk + half * 2);
        // B: 4x16 tile = K^T chunk: B[kidx][n] = K[j0+n][kk+kidx]
        v2f bb = *(const v2f*)(kb + (size_t)l * DD + kk + half * 2);
        c = __builtin_amdgcn_wmma_f32_16x16x4_f32(false, a, false, bb,
                                                  (short)0, c, false, false);
    }

    // C/D layout: VGPR r -> row r (lanes 0-15) / row r+8 (lanes 16-31), col = l
    float* __restrict__ out = attn + ((size_t)b * NN + i0) * NN + j0;
#pragma unroll
    for (int r = 0; r < 8; ++r) {
        out[(size_t)(r + half * 8) * NN + l] = c[r];
    }
}

// ---------------------------------------------------------------------------
// Kernel 2: per-row  s[j] = (S0[j] + q_row . rel_q[i,j,:]) * scale, mask,
// softmax over j, in place.  One 256-thread block per (b,i); thread t = j.
// Streams 512 MB of rel_q (read-once -> non-temporal loads, keeps the
// 192 MB L2 free for v / attn / q / k which do have reuse).
// ---------------------------------------------------------------------------
__global__ void relq_softmax_kernel(const float* __restrict__ q,
                                    const float* __restrict__ rel_q,
                                    const uint8_t* __restrict__ mask,
                                    float* __restrict__ attn) {
    const int b = blockIdx.y;
    const int i = blockIdx.x;
    const int t = threadIdx.x;            // j = t
    const int wv = t >> 5, ln = t & 31;

    __shared__ float qrow[DD];
    __shared__ float red[8];

    if (t < DD) qrow[t] = q[((size_t)b * NN + i) * DD + t];
    __syncthreads();

    const v4f* __restrict__ rq4 =
        (const v4f*)(rel_q + (((size_t)b * NN + i) * NN + t) * DD);

    float dot = 0.0f;
#pragma unroll
    for (int d4 = 0; d4 < DD / 4; ++d4) {
        v4f r4 = __builtin_nontemporal_load(rq4 + d4);   // th:NT stream
        dot += r4[0] * qrow[d4 * 4 + 0];
        dot += r4[1] * qrow[d4 * 4 + 1];
        dot += r4[2] * qrow[d4 * 4 + 2];
        dot += r4[3] * qrow[d4 * 4 + 3];
    }

    const size_t sidx = ((size_t)b * NN + i) * NN + t;
    float s = (attn[sidx] + dot) * 0.125f;   // 1/sqrt(64)
    if (__builtin_nontemporal_load(mask + sidx)) s = NEG_INF;

    // row max: intra-wave shuffle tree, then cross-wave via LDS
    float m = s;
#pragma unroll
    for (int off = 16; off > 0; off >>= 1)
        m = fmaxf(m, __shfl_xor(m, off, 32));
    if (ln == 0) red[wv] = m;
    __syncthreads();
    float mall = red[0];
#pragma unroll
    for (int w = 1; w < 8; ++w) mall = fmaxf(mall, red[w]);
    __syncthreads();                          // done reading red before reuse

    float e = __expf(s - mall);
    float sum = e;
#pragma unroll
    for (int off = 16; off > 0; off >>= 1)
        sum += __shfl_xor(sum, off, 32);
    if (ln == 0) red[wv] = sum;
    __syncthreads();
    float tot = red[0];
#pragma unroll
    for (int w = 1; w < 8; ++w) tot += red[w];

    attn[sidx] = e / tot;
}

// ---------------------------------------------------------------------------
// Kernel 3: ctx = attention @ v  per batch, via V_WMMA_F32_16X16X4_F32.
// One wave per 16x16 ctx tile (16 i-tiles x 4 d-tiles per batch).
// grid = BH*64/8 = 256 blocks of 8 waves.  v (64 KB/batch) stays L2-hot.
// ---------------------------------------------------------------------------
__global__ void av_wmma_kernel(const float* __restrict__ attn,
                               const float* __restrict__ v,
                               float* __restrict__ ctx) {
    const int wave = threadIdx.x >> 5;
    const int lane = threadIdx.x & 31;
    const int half = lane >> 4;
    const int l    = lane & 15;

    const int blk  = blockIdx.x;
    const int b    = blk >> 3;
    const int tile = (blk & 7) * 8 + wave;   // 0..63
    const int i0   = (tile >> 2) * 16;
    const int d0   = (tile & 3) * 16;

    const float* __restrict__ Ab = attn + ((size_t)b * NN + i0) * NN;
    const float* __restrict__ Vb = v + (size_t)b * NN * DD;

    v8f c = {};
#pragma unroll 8
    for (int kk = 0; kk < NN; kk += 4) {
        v2f a = *(const v2f*)(Ab + (size_t)l * NN + kk + half * 2);
        v2f bb;
        bb[0] = Vb[(size_t)(kk + half * 2 + 0) * DD + d0 + l];
        bb[1] = Vb[(size_t)(kk + half * 2 + 1) * DD + d0 + l];
        c = __builtin_amdgcn_wmma_f32_16x16x4_f32(false, a, false, bb,
                                                  (short)0, c, false, false);
    }

    float* __restrict__ out = ctx + ((size_t)b * NN + i0) * DD + d0;
#pragma unroll
    for (int r = 0; r < 8; ++r) {
        out[(size_t)(r + half * 8) * DD + l] = c[r];
    }
}

// ---------------------------------------------------------------------------
// Kernel 4: ctx[b,i,d] += sum_j attn[b,i,j] * rel_v[b,i,j,d].
// One 256-thread block per (b,i): lane d = t&63 (coalesced over rel_v),
// 4 j-groups accumulate partials, LDS-reduced.  Streams 512 MB of rel_v
// (read-once -> non-temporal loads).
// ---------------------------------------------------------------------------
__global__ void relv_ctx_kernel(const float* __restrict__ rel_v,
                                const float* __restrict__ attn,
                                float* __restrict__ ctx) {
    const int b = blockIdx.y;
    const int i = blockIdx.x;
    const int t = threadIdx.x;
    const int d = t & 63;
    const int g = t >> 6;        // 0..3

    __shared__ float arow[NN];
    __shared__ float part[4][DD];

    arow[t] = attn[((size_t)b * NN + i) * NN + t];
    __syncthreads();

    const float* __restrict__ rv = rel_v + ((size_t)b * NN + i) * NN * DD;

    float acc = 0.0f;
    const int j0 = g * 64;
#pragma unroll 8
    for (int j = j0; j < j0 + 64; ++j) {
        acc += arow[j] * __builtin_nontemporal_load(rv + (size_t)j * DD + d);
    }
    part[g][d] = acc;
    __syncthreads();

    if (g == 0) {
        float s = part[0][d] + part[1][d] + part[2][d] + part[3][d];
        ctx[((size_t)b * NN + i) * DD + d] += s;
    }
}

// ---------------------------------------------------------------------------
extern "C" void kernel_launch(void* const* d_in, const int* in_sizes, int n_in,
                              void* d_out, int out_size, void* d_ws, size_t ws_size,
                              hipStream_t stream) {
    const float*   q     = (const float*)d_in[0];
    const float*   k     = (const float*)d_in[1];
    const float*   v     = (const float*)d_in[2];
    const uint8_t* mask  = (const uint8_t*)d_in[3];   // jnp.bool_ = 1 byte
    const float*   rel_q = (const float*)d_in[4];
    const float*   rel_v = (const float*)d_in[5];

    float* ctx  = (float*)d_out;                        // [32,256,64]
    float* attn = (float*)d_out + (size_t)BH * NN * DD; // [32,256,256]

    // 1) raw q.k^T scores into the attention output region (used as scratch)
    qk_wmma_kernel<<<dim3(BH * 16 * 2), dim3(256), 0, stream>>>(q, k, attn);

    // 2) add relative-q term, scale, mask, softmax (in place)
    relq_softmax_kernel<<<dim3(NN, BH), dim3(256), 0, stream>>>(q, rel_q, mask, attn);

    // 3) ctx = attn @ v
    av_wmma_kernel<<<dim3(BH * 8), dim3(256), 0, stream>>>(attn, v, ctx);

    // 4) ctx += einsum('bij,bijd->bid', attn, rel_v)
    relv_ctx_kernel<<<dim3(NN, BH), dim3(256), 0, stream>>>(rel_v, attn, ctx);
}